// AnglesModel_57861799411905
// MI455X (gfx1250) — compile-verified
//
#include <hip/hip_runtime.h>
#include <hip/hip_bf16.h>
#include <cstdint>
#include <cstddef>

// Problem geometry (from reference): input (N_ATOMS, 3, BATCH) f32,
// angles (N_ANGLES, 3) i32, output (N_ANGLES, BATCH) f32.
#define N_ATOMS_C  2048
#define BATCH_C    512
#define TPB        128   // 4 waves; each thread owns 4 consecutive batch elems

// Native clang vector type: accepted by __builtin_nontemporal_store
// (HIP's float4 is a C++ class and is rejected).
typedef float v4f __attribute__((ext_vector_type(4)));

__global__ __launch_bounds__(TPB) void angles_cos_kernel(
    const float* __restrict__ geoms,    // (n_atoms, 3, BATCH_C)
    const int*   __restrict__ angles,   // (n_angles, 3)
    float*       __restrict__ out)      // (n_angles, BATCH_C)
{
    // 9 streams: (atom0,atom1,atom2) x (x,y,z), each BATCH_C floats.
    __shared__ __align__(16) float smem[9 * BATCH_C];   // 18 KB

    const int angle = blockIdx.x;
    const int t     = threadIdx.x;
    const int b     = t * 4;            // this thread's batch slice [b, b+4)

    // Uniform per-block gather indices (compiler emits scalar loads).
    const int a0 = angles[angle * 3 + 0];
    const int a1 = angles[angle * 3 + 1];
    const int a2 = angles[angle * 3 + 2];
    const int atoms[3] = { a0, a1, a2 };

    // Workgroup-relative LDS byte offset of smem: low 32 bits of the
    // generic (flat) address of a __shared__ object (ISA 10.2 aperture map).
    const uint32_t lds_base = (uint32_t)(uintptr_t)(&smem[0]);

    // Stage all 9 (atom, coord) 16-byte granules for this thread into LDS
    // through the CDNA5 async path (ASYNCcnt-tracked, no VGPR data).
    #pragma unroll
    for (int ai = 0; ai < 3; ++ai) {
        #pragma unroll
        for (int c = 0; c < 3; ++c) {
            const int s = ai * 3 + c;
            const uint64_t gaddr = (uint64_t)(uintptr_t)
                (geoms + ((size_t)(atoms[ai] * 3 + c) * BATCH_C + b));
            const uint32_t laddr = lds_base
                + (uint32_t)(((size_t)s * BATCH_C + b) * sizeof(float));
            asm volatile("global_load_async_to_lds_b128 %0, %1, off"
                         :: "v"(laddr), "v"(gaddr)
                         : "memory");
        }
    }

    // Each thread consumes only LDS bytes it issued itself, so the per-wave
    // async counter is a sufficient fence (no workgroup barrier needed).
    asm volatile("s_wait_asynccnt 0x0" ::: "memory");

    float res[4];
    #pragma unroll
    for (int j = 0; j < 4; ++j) {
        float dot = 0.0f, s1 = 0.0f, s2 = 0.0f;
        #pragma unroll
        for (int c = 0; c < 3; ++c) {
            const float p0 = smem[(0 * 3 + c) * BATCH_C + b + j];
            const float p1 = smem[(1 * 3 + c) * BATCH_C + b + j];
            const float p2 = smem[(2 * 3 + c) * BATCH_C + b + j];
            const float u = p0 - p1;
            const float w = p2 - p1;
            dot = fmaf(u, w, dot);
            s1  = fmaf(u, u, s1);
            s2  = fmaf(w, w, s2);
        }
        // cos = dot / (|v1||v2|) = dot * rsqrt(|v1|^2 * |v2|^2)
        res[j] = dot * rsqrtf(s1 * s2);
    }

    // 128 MB pure write stream: keep it out of the caches (NT hint) so L2
    // stays dedicated to the 12 MB reused geometry.
    v4f o = { res[0], res[1], res[2], res[3] };
    v4f* dst = reinterpret_cast<v4f*>(out + (size_t)angle * BATCH_C + b);
    __builtin_nontemporal_store(o, dst);
}

extern "C" void kernel_launch(void* const* d_in, const int* in_sizes, int n_in,
                              void* d_out, int out_size, void* d_ws, size_t ws_size,
                              hipStream_t stream) {
    (void)n_in; (void)out_size; (void)d_ws; (void)ws_size;
    const float* geoms  = (const float*)d_in[0];
    const int*   angles = (const int*)d_in[1];
    float*       out    = (float*)d_out;

    const int n_angles = in_sizes[1] / 3;   // 65536 in the reference setup

    hipLaunchKernelGGL(angles_cos_kernel,
                       dim3(n_angles), dim3(TPB), 0, stream,
                       geoms, angles, out);
}